// LogMMExpDenseSPModel_Async_32564442038610
// MI455X (gfx1250) — compile-verified
//
#include <hip/hip_runtime.h>

// Problem constants (match reference): N batch, D inner, E columns, 128 nnz/col.
constexpr int kN  = 64;
constexpr int kD  = 4096;
constexpr int kE  = 8192;
constexpr int kPC = 128;

typedef float        v2f  __attribute__((ext_vector_type(2)));
typedef float        v8f  __attribute__((ext_vector_type(8)));
typedef unsigned int u32x4 __attribute__((ext_vector_type(4)));
typedef int          i32x4 __attribute__((ext_vector_type(4)));
typedef int          i32x8 __attribute__((ext_vector_type(8)));

// Feature detection must only run on the device pass: the host pass lacks the
// amdgcn builtins (calls are deferred-diagnosed, but #if/#error are not).
#if defined(__HIP_DEVICE_COMPILE__)
#define HAVE_TDM  __has_builtin(__builtin_amdgcn_tensor_load_to_lds)
#define HAVE_WMMA __has_builtin(__builtin_amdgcn_wmma_f32_16x16x4_f32)
#else
#define HAVE_TDM  1
#define HAVE_WMMA 1
#endif

// ---------------------------------------------------------------------------
// Kernel 1: c[n] = max_d x[n,d];  PexpT[d*64+n] = exp(x[n,d] - c[n])
// 1 MB table (L2-resident), written once; main loop becomes transcendental-free.
// ---------------------------------------------------------------------------
__global__ __launch_bounds__(256) void prep_pexp_kernel(
    const float* __restrict__ x, float* __restrict__ pexpT,
    float* __restrict__ c) {
  const int n = blockIdx.x;
  const int t = threadIdx.x;
  __shared__ float red[256];
  float m = -3.402823466e+38f;
  for (int d = t; d < kD; d += 256) m = fmaxf(m, x[n * kD + d]);
  red[t] = m;
  __syncthreads();
  for (int s = 128; s > 0; s >>= 1) {
    if (t < s) red[t] = fmaxf(red[t], red[t + s]);
    __syncthreads();
  }
  const float cn = red[0];
  if (t == 0) c[n] = cn;
  for (int d = t; d < kD; d += 256)
    pexpT[d * kN + n] = expf(x[n * kD + d] - cn);
}

// ---------------------------------------------------------------------------
// Kernel 2: one wave per column e.
//   s[n] = sum_k exp(values[k]-g_e) * PexpT[rows[k]][n]   (WMMA, ones-A)
//   out[n,e] = c[n] + g_e + log(s[n])
// TDM gather mode DMAs 16 indexed rows/descriptor into LDS (double-buffered,
// TENSORcnt-pipelined); WMMA consumes from LDS. 4 waves (4 columns) / block.
// ---------------------------------------------------------------------------
__global__ __launch_bounds__(128) void lse_wmma_kernel(
    const float* __restrict__ values, const int* __restrict__ rows,
    const float* __restrict__ pexpT, const float* __restrict__ c,
    float* __restrict__ out) {
  const int lane = threadIdx.x & 31;
  const int w    = threadIdx.x >> 5;        // wave id in block (0..3)
  const int e    = blockIdx.x * 4 + w;      // column id (wave-uniform)
  if (e >= kE) return;

  __shared__ float s_a[4][kPC];
#if HAVE_TDM
  __shared__ float s_tile[4][2][16 * kN];   // per-wave double buffer, 2 x 4KB
#else
  __shared__ int s_r[4][kPC];
#endif

  // ---- per-column weights: g = max(values), a_k = exp(v_k - g) ----
  const int base = e * kPC;
  const float v0 = values[base + lane];
  const float v1 = values[base + lane + 32];
  const float v2 = values[base + lane + 64];
  const float v3 = values[base + lane + 96];
  float g = fmaxf(fmaxf(v0, v1), fmaxf(v2, v3));
  for (int off = 16; off > 0; off >>= 1) g = fmaxf(g, __shfl_xor(g, off));
  s_a[w][lane]      = expf(v0 - g);
  s_a[w][lane + 32] = expf(v1 - g);
  s_a[w][lane + 64] = expf(v2 - g);
  s_a[w][lane + 96] = expf(v3 - g);
#if !HAVE_TDM
  s_r[w][lane]      = rows[base + lane];
  s_r[w][lane + 32] = rows[base + lane + 32];
  s_r[w][lane + 64] = rows[base + lane + 64];
  s_r[w][lane + 96] = rows[base + lane + 96];
#endif
  __syncthreads();

  const int sel = lane >> 4;  // half-wave -> K slot group
  const int nn  = lane & 15;  // N index within tile

  v8f acc[4] = {};            // 4 tiles of 16 batch rows -> N = 64
  const v2f ones = {1.0f, 1.0f};

#if HAVE_WMMA && HAVE_TDM
  // ======================= TDM gather-mode path ============================
  const unsigned lds0 = __builtin_amdgcn_readfirstlane(
      (unsigned)(unsigned long long)&s_tile[w][0][0]);
  const unsigned lds1 = __builtin_amdgcn_readfirstlane(
      (unsigned)(unsigned long long)&s_tile[w][1][0]);
  const unsigned long long gbase = (unsigned long long)pexpT;

  // D# group1 (constant): data_size=4B; tensor_dim0=64; tensor_dim1=4096 rows;
  // tile_dim0=64; tile_dim1=16 gather indices; dim0 stride = 64 elements.
  i32x8 g1;
  g1[0] = 0x00020000;            // workgroup_mask=0, data_size=2 (4B)
  g1[1] = (int)(64u << 16);      // tensor_dim0[15:0] in high half
  g1[2] = (int)(4096u << 16);    // tensor_dim1[15:0] in high half
  g1[3] = (int)(64u << 16);      // tile_dim0
  g1[4] = 16;                    // tile_dim1 = #valid indices
  g1[5] = 64;                    // tensor_dim0_stride[31:0]
  g1[6] = 0;
  g1[7] = 0;

  u32x4 g0c;
  g0c[0] = 0x80000001u;          // count=1 | gather_mode=1 | 16-bit indices
  g0c[1] = 0;                    // lds_addr (filled per issue)
  g0c[2] = (unsigned)(gbase & 0xffffffffull);
  g0c[3] = (unsigned)((gbase >> 32) & 0x01ffffffull) | (2u << 30);  // type=2

  const i32x8 gx = {};           // extra group of the 6-arg form (zero-filled)

  const int* rp = rows + base;
  auto issue = [&](int grp, unsigned ldsaddr) {
    u32x4 g0 = g0c;
    g0[1] = ldsaddr;
    i32x4 g2, g3;
#pragma unroll
    for (int j = 0; j < 4; ++j) {  // pack 16-bit row indices (uniform values)
      g2[j] = __builtin_amdgcn_readfirstlane(
          (rp[grp * 16 + 2 * j] & 0xffff) | (rp[grp * 16 + 2 * j + 1] << 16));
      g3[j] = __builtin_amdgcn_readfirstlane(
          (rp[grp * 16 + 8 + 2 * j] & 0xffff) |
          (rp[grp * 16 + 9 + 2 * j] << 16));
    }
    __builtin_amdgcn_tensor_load_to_lds(g0, g1, g2, g3, gx, 0);
  };

  issue(0, lds0);
#pragma unroll
  for (int grp = 0; grp < 8; ++grp) {
    if (grp < 7) {
      issue(grp + 1, (grp & 1) ? lds0 : lds1);
      __builtin_amdgcn_s_wait_tensorcnt(1);   // previous DMA done
    } else {
      __builtin_amdgcn_s_wait_tensorcnt(0);   // last DMA done
    }
    asm volatile("" ::: "memory");            // keep LDS reads below the wait
    const float* tb = (grp & 1) ? &s_tile[w][1][0] : &s_tile[w][0][0];
#pragma unroll
    for (int kc = 0; kc < 16; kc += 4) {
      const float a0 = s_a[w][grp * 16 + kc + sel];
      const float a1 = s_a[w][grp * 16 + kc + 2 + sel];
      const float* p0 = tb + (kc + sel) * kN + nn;
      const float* p1 = tb + (kc + 2 + sel) * kN + nn;
#pragma unroll
      for (int t = 0; t < 4; ++t) {
        v2f b;                      // weight folded in: K-slot permutation-safe
        b.x = a0 * p0[t * 16];
        b.y = a1 * p1[t * 16];
        acc[t] = __builtin_amdgcn_wmma_f32_16x16x4_f32(
            false, ones, false, b, (short)0, acc[t], false, false);
      }
    }
  }
#elif HAVE_WMMA
  // ================== direct L2-gather path (no TDM builtin) ===============
  for (int kc = 0; kc < kPC; kc += 4) {
    const float a0 = s_a[w][kc + sel];
    const float a1 = s_a[w][kc + 2 + sel];
    const int   r0 = s_r[w][kc + sel];
    const int   r1 = s_r[w][kc + 2 + sel];
    const float* __restrict__ p0 = pexpT + r0 * kN + nn;
    const float* __restrict__ p1 = pexpT + r1 * kN + nn;
#pragma unroll
    for (int t = 0; t < 4; ++t) {
      v2f b;
      b.x = a0 * p0[t * 16];
      b.y = a1 * p1[t * 16];
      acc[t] = __builtin_amdgcn_wmma_f32_16x16x4_f32(
          false, ones, false, b, (short)0, acc[t], false, false);
    }
  }
#else
#error "no f32 WMMA builtin on this device toolchain"
#endif

  // D layout: VGPR0, lanes 0-15 hold row M=0, N=lane; all D rows identical
  // (ones-A), so lanes 0..15 carry s[n] for each tile.
#pragma unroll
  for (int t = 0; t < 4; ++t) {
    if (lane < 16) {
      const int n = t * 16 + nn;
      out[(size_t)n * kE + e] = c[n] + g + logf(acc[t][0]);
    }
  }
}

// ---------------------------------------------------------------------------
extern "C" void kernel_launch(void* const* d_in, const int* in_sizes, int n_in,
                              void* d_out, int out_size, void* d_ws,
                              size_t ws_size, hipStream_t stream) {
  const float* x      = (const float*)d_in[0];   // [N, D]
  const float* values = (const float*)d_in[1];   // [NNZ]
  const int*   rows   = (const int*)d_in[2];     // [NNZ]
  (void)d_in[3];  // col_ids implicit: column e owns [e*128, (e+1)*128)
  (void)in_sizes; (void)n_in; (void)out_size; (void)ws_size;

  float* out   = (float*)d_out;                  // [N, E]
  float* pexpT = (float*)d_ws;                   // [D, N]  (1 MB)
  float* c     = pexpT + (size_t)kD * kN;        // [N]

  prep_pexp_kernel<<<kN, 256, 0, stream>>>(x, pexpT, c);
  lse_wmma_kernel<<<kE / 4, 128, 0, stream>>>(values, rows, pexpT, c, out);
}